// tNN4MNIST_33054068310095
// MI455X (gfx1250) — compile-verified
//
#include <hip/hip_runtime.h>
#include <math.h>

typedef float v2f __attribute__((ext_vector_type(2)));
typedef float v4f __attribute__((ext_vector_type(4)));
typedef float v8f __attribute__((ext_vector_type(8)));

#define NTUBE  28
#define NCOLS  65536
#define NT     16            // n-columns per workgroup tile
#define PITCH  512           // LDS row pitch in floats (32 rows x 512 cols)
#define NWAVES 8
#define PI_F   3.14159265358979323846f

#define ACT_FLOATS (28*28*65536)   // 51,380,224 floats = 205.5 MB

// D[f][l] = 2*cos(pi*f*(2l+1)/(2*28)), zero-padded to 32x32
__device__ __forceinline__ float dct_D(int f, int l) {
    if (f >= NTUBE || l >= NTUBE) return 0.0f;
    return 2.0f * cosf(PI_F * (float)f * (float)(2 * l + 1) / (2.0f * NTUBE));
}
// Exact inverse of D (closed form for unnormalized DCT-II):
// Dinv[l][f] = (1/28) * (f==0 ? 0.5 : cos(pi*f*(2l+1)/56)), zero-padded
__device__ __forceinline__ float dct_Dinv(int l, int f) {
    if (f >= NTUBE || l >= NTUBE) return 0.0f;
    float c = (f == 0) ? 1.0f
                       : 2.0f * cosf(PI_F * (float)f * (float)(2 * l + 1) / (2.0f * NTUBE));
    return c * (1.0f / 56.0f);
}

// Build padded D (1024 floats) and Dinv (1024 floats) tables once. <<<8,256>>>
__global__ void tab_init(float* __restrict__ dtab) {
    int e = blockIdx.x * blockDim.x + threadIdx.x;
    if (e < 1024)      dtab[e] = dct_D(e >> 5, e & 31);
    else if (e < 2048) { int i = e - 1024; dtab[e] = dct_Dinv(i >> 5, i & 31); }
}

// Precompute Wh[f][m][k] = sum_l D[f][l] * W[l][m][k], padded to 32 x 32x32.
// grid.x = 32 (frequency, incl. zero pad rows), 256 threads.
template <int MW>
__global__ void wh_init(const float* __restrict__ W, float* __restrict__ Wh) {
    const int f = blockIdx.x;
    for (int e = threadIdx.x; e < 32 * 32; e += blockDim.x) {
        int m = e >> 5, k = e & 31;
        float v = 0.0f;
        if (f < NTUBE && m < MW && k < NTUBE) {
            #pragma unroll
            for (int l = 0; l < NTUBE; ++l)
                v += dct_D(f, l) * W[(l * MW + m) * NTUBE + k];
        }
        Wh[f * 1024 + e] = v;
    }
}

// One t-product layer, fully fused: DCT -> per-frequency GEMM -> IDCT -> bias(+ReLU).
// in : (28, 28, 65536) f32    out : (28, MOUT, 65536) f32
// Wh : 32 slices of padded 32x32 f32 (rows f>=28 are zero)
template <int MOUT, bool RELU>
__global__ void __launch_bounds__(256) tlayer(
    const float* __restrict__ in, float* __restrict__ out,
    const float* __restrict__ Wh, const float* __restrict__ bias,
    const float* __restrict__ dtab)
{
    extern __shared__ float lds[];
    float* bufA = lds;                    // 32*512: X tile, then Yh
    float* bufB = lds + 32 * PITCH;       // 32*512: Xh, then Y
    float* Dl   = lds + 2 * 32 * PITCH;   // 1024:  D[f][l]   (padded)
    float* Di   = Dl + 1024;              // 1024:  Dinv[l][f] (padded)

    const int tid  = threadIdx.x;
    const int lane = tid & 31;
    // Wave index as a true SGPR so job loops are scalar (EXEC stays all-1s,
    // which WMMA requires) instead of exec-masked vector loops.
    const int wave = __builtin_amdgcn_readfirstlane(tid >> 5);
    const int lo   = lane & 15;           // M/N position within a 16-wide tile
    const int hi   = lane >> 4;           // half-wave selector for K striping
    const int n0   = blockIdx.x * NT;

    // ---- Copy DCT tables into LDS (2048 floats = 512 float4 chunks) --------
    for (int e = tid; e < 512; e += 256)
        *(v4f*)(Dl + 4 * e) = *(const v4f*)(dtab + 4 * e);
    // Zero pad columns 448..511 of bufB (k = 28..31 of Xh read by stage B)
    for (int e = tid; e < 32 * 16; e += 256)
        *(v4f*)(bufB + (e >> 4) * PITCH + 448 + 4 * (e & 15)) = (v4f){0.f, 0.f, 0.f, 0.f};
    // ---- Load X tile as float4: bufA[l][k*16+n], rows l=28..31 zeroed ------
    for (int e = tid; e < 32 * 28 * 4; e += 256) {   // 14 iterations exactly
        int l = e / (28 * 4);
        int r = e % (28 * 4);
        int k = r >> 2, c = r & 3;
        v4f v = {0.f, 0.f, 0.f, 0.f};
        if (l < NTUBE) v = *(const v4f*)(in + (l * NTUBE + k) * NCOLS + n0 + 4 * c);
        *(v4f*)(bufA + l * PITCH + k * 16 + 4 * c) = v;
    }
    __syncthreads();

    // ---- Stage A: Xh(bufB) = D~(32x32) @ X(bufA, 32 x 448) -----------------
    // 2 M-tiles (f) x 28 N-tiles (k,n) jobs, K = 32 (8 wmma steps)
    for (int j = wave; j < 2 * 28; j += NWAVES) {
        int mt = j & 1, nt = j >> 1;
        v8f acc = {0.f, 0.f, 0.f, 0.f, 0.f, 0.f, 0.f, 0.f};
        #pragma unroll
        for (int ks = 0; ks < 8; ++ks) {
            v2f a, b;
            a.x = Dl[(mt * 16 + lo) * 32 + ks * 4 + 2 * hi + 0];
            a.y = Dl[(mt * 16 + lo) * 32 + ks * 4 + 2 * hi + 1];
            b.x = bufA[(ks * 4 + 0 + 2 * hi) * PITCH + nt * 16 + lo];
            b.y = bufA[(ks * 4 + 1 + 2 * hi) * PITCH + nt * 16 + lo];
            acc = __builtin_amdgcn_wmma_f32_16x16x4_f32(
                false, a, false, b, (short)0, acc, false, false);
        }
        #pragma unroll
        for (int i = 0; i < 8; ++i)
            bufB[(mt * 16 + i + 8 * hi) * PITCH + nt * 16 + lo] = acc[i];
    }
    __syncthreads();

    // ---- Stage B: per-frequency Yh[f](bufA) = Wh[f](32x32) @ Xh[f](32x16) --
    constexpr int MTB = (MOUT + 15) >> 4;     // 2 for MOUT=28, 1 for MOUT=10
    for (int j = wave; j < 32 * MTB; j += NWAVES) {
        int f = j / MTB, mt = j % MTB;
        const float* __restrict__ Whf = Wh + f * 1024;
        v8f acc = {0.f, 0.f, 0.f, 0.f, 0.f, 0.f, 0.f, 0.f};
        #pragma unroll
        for (int ks = 0; ks < 8; ++ks) {
            v2f a, b;
            a.x = Whf[(mt * 16 + lo) * 32 + ks * 4 + 2 * hi + 0];
            a.y = Whf[(mt * 16 + lo) * 32 + ks * 4 + 2 * hi + 1];
            b.x = bufB[f * PITCH + (ks * 4 + 0 + 2 * hi) * 16 + lo];
            b.y = bufB[f * PITCH + (ks * 4 + 1 + 2 * hi) * 16 + lo];
            acc = __builtin_amdgcn_wmma_f32_16x16x4_f32(
                false, a, false, b, (short)0, acc, false, false);
        }
        #pragma unroll
        for (int i = 0; i < 8; ++i)
            bufA[f * PITCH + (mt * 16 + i + 8 * hi) * 16 + lo] = acc[i];
    }
    __syncthreads();

    // ---- Stage C: Y(bufB) = Dinv~(32x32) @ Yh(bufA, 32 x MOUT*16) ----------
    // N-tile nt corresponds exactly to output channel m (NT == 16)
    for (int j = wave; j < 2 * MOUT; j += NWAVES) {
        int mt = j & 1, nt = j >> 1;
        v8f acc = {0.f, 0.f, 0.f, 0.f, 0.f, 0.f, 0.f, 0.f};
        #pragma unroll
        for (int ks = 0; ks < 8; ++ks) {
            v2f a, b;
            a.x = Di[(mt * 16 + lo) * 32 + ks * 4 + 2 * hi + 0];
            a.y = Di[(mt * 16 + lo) * 32 + ks * 4 + 2 * hi + 1];
            b.x = bufA[(ks * 4 + 0 + 2 * hi) * PITCH + nt * 16 + lo];
            b.y = bufA[(ks * 4 + 1 + 2 * hi) * PITCH + nt * 16 + lo];
            acc = __builtin_amdgcn_wmma_f32_16x16x4_f32(
                false, a, false, b, (short)0, acc, false, false);
        }
        #pragma unroll
        for (int i = 0; i < 8; ++i)
            bufB[(mt * 16 + i + 8 * hi) * PITCH + nt * 16 + lo] = acc[i];
    }
    __syncthreads();

    // ---- Epilogue: bias + ReLU + float4 store ------------------------------
    for (int e = tid; e < NTUBE * MOUT * 4; e += 256) {
        int l = e / (MOUT * 4);
        int r = e % (MOUT * 4);
        int m = r >> 2, c = r & 3;
        float bv = bias[l * MOUT + m];
        v4f v = *(const v4f*)(bufB + l * PITCH + m * 16 + 4 * c);
        v.x += bv; v.y += bv; v.z += bv; v.w += bv;
        if (RELU) {
            v.x = fmaxf(v.x, 0.0f); v.y = fmaxf(v.y, 0.0f);
            v.z = fmaxf(v.z, 0.0f); v.w = fmaxf(v.w, 0.0f);
        }
        *(v4f*)(out + (l * MOUT + m) * NCOLS + n0 + 4 * c) = v;
    }
}

extern "C" void kernel_launch(void* const* d_in, const int* in_sizes, int n_in,
                              void* d_out, int out_size, void* d_ws, size_t ws_size,
                              hipStream_t stream) {
    (void)in_sizes; (void)n_in; (void)out_size; (void)ws_size;
    const float* x  = (const float*)d_in[0];
    const float* W1 = (const float*)d_in[1];
    const float* B1 = (const float*)d_in[2];
    const float* W2 = (const float*)d_in[3];
    const float* B2 = (const float*)d_in[4];
    const float* W3 = (const float*)d_in[5];
    const float* B3 = (const float*)d_in[6];
    const float* W4 = (const float*)d_in[7];
    const float* B4 = (const float*)d_in[8];
    float* out = (float*)d_out;

    float* act  = (float*)d_ws;              // 205.5 MB activation ping buffer
    float* wh1  = act + ACT_FLOATS;          // 4 x 32*32*32 f32 = 512 KB total
    float* wh2  = wh1 + 32 * 1024;
    float* wh3  = wh2 + 32 * 1024;
    float* wh4  = wh3 + 32 * 1024;
    float* dtab = wh4 + 32 * 1024;           // 2048 f32: D~ then Dinv~

    // Tiny precompute: DCT tables + frequency-domain weights (L2-resident)
    tab_init<<<8, 256, 0, stream>>>(dtab);
    wh_init<28><<<32, 256, 0, stream>>>(W1, wh1);
    wh_init<28><<<32, 256, 0, stream>>>(W2, wh2);
    wh_init<28><<<32, 256, 0, stream>>>(W3, wh3);
    wh_init<10><<<32, 256, 0, stream>>>(W4, wh4);

    const size_t ldsBytes = (size_t)(2 * 32 * PITCH + 2 * 1024) * sizeof(float); // 136 KB
    dim3 grid(NCOLS / NT);   // 4096 workgroups, 16 columns each

    tlayer<28, true ><<<grid, 256, ldsBytes, stream>>>(x,   act, wh1, B1, dtab);
    tlayer<28, true ><<<grid, 256, ldsBytes, stream>>>(act, act, wh2, B2, dtab);  // in-place per tile
    tlayer<28, true ><<<grid, 256, ldsBytes, stream>>>(act, act, wh3, B3, dtab);
    tlayer<10, false><<<grid, 256, ldsBytes, stream>>>(act, out, wh4, B4, dtab);
}